// LeapfrogIntegrator_33363305955951
// MI455X (gfx1250) — compile-verified
//
#include <hip/hip_runtime.h>
#include <math.h>

typedef __attribute__((ext_vector_type(2))) float v2f;
typedef __attribute__((ext_vector_type(8))) float v8f;

#define MDIM   32
#define MELEMS 1024          // 32*32
#define OUTROW 1027          // K*K + 3
#define DT_F   0.01f
#define NS_ITERS 15
#define EXP_TERMS 10

// ---------------------------------------------------------------------------
// Wave-cooperative 32x32 fp32 matmul: D = A * B  (A,B,D in LDS, row-major 32)
// 4 waves per block; wave w computes output tile (w>>1, w&1).
// Uses V_WMMA_F32_16X16X4_F32, 8 chained WMMAs along K.
// ISA layouts (CDNA5 ISA §7.12.2):
//   A 16x4 fp32:  lane = M (0..15 twice), vgpr0/1 = K, lanes16-31 hold K+2
//   C/D 16x16:    vgpr v: lanes0-15 -> M=v, lanes16-31 -> M=v+8, N=lane&15
// ---------------------------------------------------------------------------
__device__ __forceinline__ void mm32(const float* __restrict__ A,
                                     const float* __restrict__ B,
                                     float* __restrict__ D,
                                     int wave, int lane) {
    const int ti  = (wave >> 1) & 1;
    const int tj  = wave & 1;
    const int row = lane & 15;
    const int kk  = (lane >> 4) << 1;            // 0 for lanes 0-15, 2 for 16-31
    v8f c = {0.f, 0.f, 0.f, 0.f, 0.f, 0.f, 0.f, 0.f};
#pragma unroll
    for (int kb = 0; kb < MDIM; kb += 4) {
        v2f a, b;
        a.x = A[(ti * 16 + row) * MDIM + kb + kk];
        a.y = A[(ti * 16 + row) * MDIM + kb + kk + 1];
        b.x = B[(kb + kk    ) * MDIM + tj * 16 + row];
        b.y = B[(kb + kk + 1) * MDIM + tj * 16 + row];
        c = __builtin_amdgcn_wmma_f32_16x16x4_f32(false, a, false, b,
                                                  (short)0, c, false, false);
    }
    const int mbase = ti * 16 + ((lane >> 4) << 3);
    const int n     = tj * 16 + row;
#pragma unroll
    for (int v = 0; v < 8; ++v)
        D[(mbase + v) * MDIM + n] = c[v];
}

__global__ void __launch_bounds__(128)
leapfrog_spd_kernel(const float* __restrict__ gSigma,
                    const float* __restrict__ gPi,
                    const float* __restrict__ gPhi,
                    const float* __restrict__ gPiPhi,
                    float* __restrict__ gOut) {
    __shared__ float S[MELEMS], P[MELEMS], T1[MELEMS], T2[MELEMS];
    __shared__ float Y[MELEMS], Z[MELEMS], W[MELEMS], E[MELEMS];
    __shared__ float red[128];
    __shared__ float scal[2];   // [0]=frob norm c, [1]=1/c

    const int m    = blockIdx.x;          // matrix index in [0, B*N)
    const int t    = threadIdx.x;         // 0..127
    const int wave = t >> 5;
    const int lane = t & 31;

    const size_t mbase = (size_t)m * MELEMS;

    // ---- load Sigma, pi_Sigma (float4 vectorized: 2 x float4 per thread) ----
    {
        const float4* Sg = (const float4*)(gSigma + mbase);
        const float4* Pg = (const float4*)(gPi + mbase);
        float4* S4 = (float4*)S;
        float4* P4 = (float4*)P;
        for (int i = t; i < MELEMS / 4; i += 128) { S4[i] = Sg[i]; P4[i] = Pg[i]; }
    }
    __syncthreads();

    // ---- momentum half-step: grad_T = sym(2 P S P); P½ = P - 0.5*DT*grad ----
    mm32(P, S, T1, wave, lane);  __syncthreads();   // T1 = P S
    mm32(T1, P, T2, wave, lane); __syncthreads();   // T2 = P S P
    for (int i = t; i < MELEMS; i += 128) {
        int r = i >> 5, c = i & 31;
        P[i] -= DT_F * 0.5f * (T2[i] + T2[c * MDIM + r]);   // DT*sym(T2)
    }
    __syncthreads();

    // ---- Sigma_dot = 2 S P½ S ; V = DT * Sigma_dot (into E) ----
    mm32(P, S, T1, wave, lane);  __syncthreads();   // T1 = P½ S
    mm32(S, T1, T2, wave, lane); __syncthreads();   // T2 = S P½ S

    // E = 2*DT*T2 ; S := sym(S) + 1e-8 I (two-phase to avoid LDS race)
    {
        float symreg[8];
        int c0 = 0;
        for (int i = t; i < MELEMS; i += 128) {
            int r = i >> 5, c = i & 31;
            symreg[c0++] = 0.5f * (S[i] + S[c * MDIM + r]);
            E[i] = 2.0f * DT_F * T2[i];
        }
        __syncthreads();
        c0 = 0;
        for (int i = t; i < MELEMS; i += 128)
            S[i] = symreg[c0++] + ((i % 33) == 0 ? 1e-8f : 0.0f);
    }
    __syncthreads();

    // ---- Frobenius norm of S for Newton-Schulz normalization ----
    {
        float s = 0.f;
        for (int i = t; i < MELEMS; i += 128) s += S[i] * S[i];
        red[t] = s;
        __syncthreads();
        for (int off = 64; off > 0; off >>= 1) {
            if (t < off) red[t] += red[t + off];
            __syncthreads();
        }
        if (t == 0) {
            float c = sqrtf(red[0]);
            scal[0] = c;
            scal[1] = 1.0f / c;
        }
        __syncthreads();
    }
    const float cnorm  = scal[0];
    const float rcnorm = scal[1];

    // ---- coupled Newton-Schulz: Ya->S^{1/2}/sqrt(c), Za->S^{-1/2}*sqrt(c) ----
    // Ping-pong between (Y,Z) and (S,P): S/P are dead from here on, so reuse
    // them as the second buffer pair (no per-iteration copy-back, one barrier
    // per iteration saved).
    for (int i = t; i < MELEMS; i += 128) {
        Y[i] = S[i] * rcnorm;                       // Y0 = S / c (reads old S)
    }
    __syncthreads();                                // Y0 written before Z init
    for (int i = t; i < MELEMS; i += 128)
        Z[i] = ((i % 33) == 0) ? 1.0f : 0.0f;       // Z0 = I
    __syncthreads();

    float* Ya = Y;  float* Za = Z;                  // current iterates
    float* Yb = S;  float* Zb = P;                  // next iterates
    for (int it = 0; it < NS_ITERS; ++it) {
        mm32(Za, Ya, T1, wave, lane);               // T1 = Z Y
        __syncthreads();
        for (int i = t; i < MELEMS; i += 128)
            T2[i] = ((i % 33) == 0 ? 1.5f : 0.0f) - 0.5f * T1[i];  // (3I-ZY)/2
        __syncthreads();
        mm32(Ya, T2, Yb, wave, lane);               // newY = Y T   (independent)
        mm32(T2, Za, Zb, wave, lane);               // newZ = T Z   (independent)
        __syncthreads();
        float* tmp;
        tmp = Ya; Ya = Yb; Yb = tmp;
        tmp = Za; Za = Zb; Zb = tmp;
    }

    // ---- W = sym(S^{-1/2} V S^{-1/2}) = (1/c) sym(Z V Z)  (V in E) ----
    // eigenvalue clip to +/-20 is a provable no-op here (||W|| ~ 0.1)
    mm32(Za, E, T1, wave, lane);  __syncthreads();
    mm32(T1, Za, T2, wave, lane); __syncthreads();
    for (int i = t; i < MELEMS; i += 128) {
        int r = i >> 5, c = i & 31;
        W[i] = rcnorm * 0.5f * (T2[i] + T2[c * MDIM + r]);
    }
    __syncthreads();

    // ---- E = exp(W) via Horner-form Taylor (||W|| small => rapid conv.) ----
    for (int i = t; i < MELEMS; i += 128)
        E[i] = ((i % 33) == 0 ? 1.0f : 0.0f) + W[i] * (1.0f / (float)EXP_TERMS);
    __syncthreads();
    for (int k = EXP_TERMS - 1; k >= 1; --k) {
        mm32(W, E, T1, wave, lane); __syncthreads();
        const float inv = 1.0f / (float)k;
        for (int i = t; i < MELEMS; i += 128)
            E[i] = ((i % 33) == 0 ? 1.0f : 0.0f) + T1[i] * inv;
        __syncthreads();
    }

    // ---- R = S^{1/2} expW S^{1/2} = c * (Y E Y); ensure_spd -> sym + 1e-6 I
    mm32(Ya, E, T1, wave, lane);  __syncthreads();
    mm32(T1, Ya, T2, wave, lane); __syncthreads();

    float* out = gOut + (size_t)m * OUTROW;
    for (int i = t; i < MELEMS; i += 128) {
        int r = i >> 5, c = i & 31;
        out[i] = cnorm * 0.5f * (T2[i] + T2[c * MDIM + r]) +
                 ((i % 33) == 0 ? 1e-6f : 0.0f);
    }

    // ---- gauge field update + mod2pi retraction (3 floats per matrix) ----
    if (t == 0) {
        const float pi_f    = 3.14159265358979323846f;
        const float two_pi  = 6.28318530717958647692f;
        const float r_max   = pi_f - 0.01f;
        float p0 = gPhi[3 * m + 0] + DT_F * gPiPhi[3 * m + 0];
        float p1 = gPhi[3 * m + 1] + DT_F * gPiPhi[3 * m + 1];
        float p2 = gPhi[3 * m + 2] + DT_F * gPiPhi[3 * m + 2];
        float theta = sqrtf(p0 * p0 + p1 * p1 + p2 * p2);
        float tsafe = fmaxf(theta, 1e-12f);
        float a0 = p0 / tsafe, a1 = p1 / tsafe, a2 = p2 / tsafe;
        float tw = fmodf(theta, two_pi);
        bool  flip = tw > pi_f;
        float tf = flip ? (two_pi - tw) : tw;
        float sg = flip ? -1.0f : 1.0f;
        tf = fminf(tf, r_max);
        out[MELEMS + 0] = a0 * sg * tf;
        out[MELEMS + 1] = a1 * sg * tf;
        out[MELEMS + 2] = a2 * sg * tf;
    }
}

extern "C" void kernel_launch(void* const* d_in, const int* in_sizes, int n_in,
                              void* d_out, int out_size, void* d_ws, size_t ws_size,
                              hipStream_t stream) {
    (void)n_in; (void)out_size; (void)d_ws; (void)ws_size;
    const float* Sigma  = (const float*)d_in[0];
    const float* piS    = (const float*)d_in[1];
    const float* phi    = (const float*)d_in[2];
    const float* piphi  = (const float*)d_in[3];
    float* out = (float*)d_out;

    const int nmat = in_sizes[0] / MELEMS;   // B*N = 32768
    leapfrog_spd_kernel<<<nmat, 128, 0, stream>>>(Sigma, piS, phi, piphi, out);
}